// PartitionedNormalization_86955907875125
// MI455X (gfx1250) — compile-verified
//
#include <hip/hip_runtime.h>

#define FEAT          1024
#define EPS           1e-5f
#define THREADS       256
#define ROWS_PER_BLK  16
#define NWAVES        (THREADS / 32)
#define NBUF          3   // triple-buffer LDS row staging

typedef int v4i_t __attribute__((ext_vector_type(4)));
typedef __attribute__((address_space(1))) void  as1_void;
typedef __attribute__((address_space(3))) void  as3_void;
typedef __attribute__((address_space(1))) v4i_t as1_v4i;
typedef __attribute__((address_space(3))) v4i_t as3_v4i;

// ---- gfx1250 async global->LDS copy (ASYNCcnt path) ----------------------
__device__ __forceinline__ void async_load_b128(const float* gsrc, float* ldst) {
#if __has_builtin(__builtin_amdgcn_global_load_async_to_lds_b128)
    __builtin_amdgcn_global_load_async_to_lds_b128(
        (as1_v4i*)(as1_void*)gsrc, (as3_v4i*)(as3_void*)ldst,
        /*offset=*/0, /*cpol=*/0);
#else
    unsigned           loff  = (unsigned)(unsigned long long)(as3_void*)ldst;
    unsigned long long gaddr = (unsigned long long)gsrc;
    asm volatile("global_load_async_to_lds_b128 %0, %1, off"
                 :: "v"(loff), "v"(gaddr) : "memory");
#endif
}

template <int N>
__device__ __forceinline__ void wait_asynccnt() {
#if __has_builtin(__builtin_amdgcn_s_wait_asynccnt)
    __builtin_amdgcn_s_wait_asynccnt(N);
#else
    asm volatile("s_wait_asynccnt %0" :: "i"(N) : "memory");
#endif
}

// ---- wave32 XOR-butterfly step via DS_SWIZZLE_B32 (group-of-32 mode) -----
// offset encoding: [15]=0, xor_mask<<10 | or_mask<<5 | and_mask(0x1f)
template <int XOR_MASK>
__device__ __forceinline__ float swz_xor_add(float v) {
    constexpr int pat = (XOR_MASK << 10) | 0x1f;
    return v + __int_as_float(__builtin_amdgcn_ds_swizzle(__float_as_int(v), pat));
}

__device__ __forceinline__ float wave_reduce(float v) {
    v = swz_xor_add<16>(v);
    v = swz_xor_add<8>(v);
    v = swz_xor_add<4>(v);
    v = swz_xor_add<2>(v);
    v = swz_xor_add<1>(v);
    return v;
}

// ---- kernel --------------------------------------------------------------
__global__ __launch_bounds__(THREADS)
void pnorm_kernel(const float* __restrict__ x,
                  const int*   __restrict__ domain_ids,
                  const float* __restrict__ weight,
                  const float* __restrict__ bias,
                  float*       __restrict__ out,
                  int rows)
{
    __shared__ float  xbuf[NBUF][FEAT];     // 12 KB row staging
    __shared__ float2 red[2][NWAVES];       // cross-wave partials, parity-buffered

    const int tid  = threadIdx.x;
    const int wid  = tid >> 5;
    const int lane = tid & 31;
    const int col  = tid << 2;              // 4 floats per thread, FEAT = THREADS*4

    const long long row0 = (long long)blockIdx.x * ROWS_PER_BLK;
    const long long last = (long long)rows - 1;

    // prologue: prefetch first row into buffer 0
    {
        long long r0 = row0 <= last ? row0 : last;
        async_load_b128(x + r0 * FEAT + col, &xbuf[0][col]);
    }

    int cur = 0;
    for (int r = 0; r < ROWS_PER_BLK; ++r) {
        const long long row = row0 + r;
        const int nxt = (cur + 1 == NBUF) ? 0 : cur + 1;

        if (r + 1 < ROWS_PER_BLK) {
            long long prow = (row + 1) <= last ? (row + 1) : last;
            async_load_b128(x + prow * FEAT + col, &xbuf[nxt][col]);
            wait_asynccnt<1>();             // in-order: row r has landed in LDS
        } else {
            wait_asynccnt<0>();
        }

        // read back own 16B chunk (ds_load_b128)
        const float4 v = *(const float4*)&xbuf[cur][col];

        float s  = v.x + v.y + v.z + v.w;
        float s2 = v.x * v.x;
        s2 = fmaf(v.y, v.y, s2);
        s2 = fmaf(v.z, v.z, s2);
        s2 = fmaf(v.w, v.w, s2);

        // wave32 butterfly reduction (ds_swizzle, no lane-clamp ALU)
        s  = wave_reduce(s);
        s2 = wave_reduce(s2);

        const int par = r & 1;
        if (lane == 0) red[par][wid] = make_float2(s, s2);
        __syncthreads();

        float ts = 0.f, ts2 = 0.f;
        #pragma unroll
        for (int w = 0; w < NWAVES; ++w) {
            float2 p = red[par][w];         // same-address broadcast reads
            ts += p.x; ts2 += p.y;
        }

        const float invN = 1.0f / (float)FEAT;
        const float mean = ts * invN;
        const float var  = fmaf(-mean, mean, ts2 * invN);   // E[x^2] - mean^2
        const float rstd = rsqrtf(var + EPS);

        if (row <= last) {
            const int dom = domain_ids[row];                // block-uniform -> s_load
            const float4 wv = *(const float4*)(weight + (long long)dom * FEAT + col);
            const float4 bv = *(const float4*)(bias   + (long long)dom * FEAT + col);

            float4 o;
            o.x = fmaf((v.x - mean) * rstd, wv.x, bv.x);
            o.y = fmaf((v.y - mean) * rstd, wv.y, bv.y);
            o.z = fmaf((v.z - mean) * rstd, wv.z, bv.z);
            o.w = fmaf((v.w - mean) * rstd, wv.w, bv.w);
            *(float4*)(out + row * FEAT + col) = o;          // global_store_b128
        }
        cur = nxt;
    }
}

// ---- host launch ---------------------------------------------------------
extern "C" void kernel_launch(void* const* d_in, const int* in_sizes, int n_in,
                              void* d_out, int out_size, void* d_ws, size_t ws_size,
                              hipStream_t stream) {
    const float* x          = (const float*)d_in[0];
    const int*   domain_ids = (const int*)  d_in[1];
    const float* weight     = (const float*)d_in[2];
    const float* bias       = (const float*)d_in[3];
    float*       out        = (float*)d_out;

    const int rows = in_sizes[1];                     // BATCH (domain_ids count)
    const int grid = (rows + ROWS_PER_BLK - 1) / ROWS_PER_BLK;

    pnorm_kernel<<<grid, THREADS, 0, stream>>>(x, domain_ids, weight, bias, out, rows);
}